// BinRegularizer_41412074668319
// MI455X (gfx1250) — compile-verified
//
#include <hip/hip_runtime.h>

// -------- types --------
typedef float v4f __attribute__((ext_vector_type(4)));
typedef float v2f __attribute__((ext_vector_type(2)));
typedef float v8f __attribute__((ext_vector_type(8)));

#define NBLK 2048
#define TPB  256
#define NACC 12   // M1 M2 M3 | W0 W1 W2 W3 | Q0 Q1 Q2 Q3 | sumAbsD

// ---------------------------------------------------------------------------
// Pass 1: streaming moment accumulation over the weight matrix.
// Accumulates Sum(b^p), Sum(b^p * w), Sum(b^p * w^2) for p=0..3 and Sum|w - a*b|
// where b = rint(clamp(w/a, -2, 1)).  One 16-float partial row per block.
// ---------------------------------------------------------------------------
__global__ __launch_bounds__(TPB) void bin_moments_kernel(
    const v4f* __restrict__ w4, const float* __restrict__ alpha,
    float* __restrict__ ws, int n4)
{
  const float a  = alpha[0];
  const float ia = 1.0f / a;

  float M1 = 0.f, M2 = 0.f, M3 = 0.f;
  float W0 = 0.f, W1 = 0.f, W2 = 0.f, W3 = 0.f;
  float Q0 = 0.f, Q1 = 0.f, Q2 = 0.f, Q3 = 0.f;
  float SA = 0.f;

  const int tid    = blockIdx.x * TPB + threadIdx.x;
  const int stride = NBLK * TPB;

  for (int i = tid; i < n4; i += stride) {
    v4f v = __builtin_nontemporal_load(&w4[i]);   // global_load_b128, TH=NT
#pragma unroll
    for (int k = 0; k < 4; ++k) {
      float w  = v[k];
      float t  = w * ia;
      float c  = __builtin_amdgcn_fmed3f(t, -2.0f, 1.0f);  // clamp in 1 op
      float b  = __builtin_rintf(c);                        // v_rndne_f32
      float d  = __builtin_fmaf(b, -a, w);                  // w - a*b
      SA += __builtin_fabsf(d);
      float w2 = w * w;
      float b2 = b * b;
      float b3 = b2 * b;
      M1 += b;  M2 += b2;  M3 += b3;
      W0 += w;
      W1 = __builtin_fmaf(b,  w, W1);
      W2 = __builtin_fmaf(b2, w, W2);
      W3 = __builtin_fmaf(b3, w, W3);
      Q0 += w2;
      Q1 = __builtin_fmaf(b,  w2, Q1);
      Q2 = __builtin_fmaf(b2, w2, Q2);
      Q3 = __builtin_fmaf(b3, w2, Q3);
    }
  }

  __shared__ float red[NACC][TPB];
  const int tx = threadIdx.x;
  red[0][tx]  = M1; red[1][tx]  = M2; red[2][tx]  = M3;
  red[3][tx]  = W0; red[4][tx]  = W1; red[5][tx]  = W2; red[6][tx] = W3;
  red[7][tx]  = Q0; red[8][tx]  = Q1; red[9][tx]  = Q2; red[10][tx] = Q3;
  red[11][tx] = SA;

  for (int s = TPB / 2; s > 0; s >>= 1) {
    __syncthreads();
    if (tx < s) {
#pragma unroll
      for (int k = 0; k < NACC; ++k) red[k][tx] += red[k][tx + s];
    }
  }

  if (tx == 0) {
#pragma unroll
    for (int k = 0; k < NACC; ++k) ws[blockIdx.x * 16 + k] = red[k][0];
#pragma unroll
    for (int k = NACC; k < 16; ++k) ws[blockIdx.x * 16 + k] = 0.0f;
  }
}

// ---------------------------------------------------------------------------
// Pass 2: reduce the (NBLK x 16) partial matrix with V_WMMA_F32_16X16X4_F32.
// A = ones(16x4)  =>  D[m][n] = sum_k B[k][n] + C[m][n]: each WMMA folds 4
// partial rows into all 16 columns. 8 waves x 64 chained WMMAs, then LDS
// combine; lane 0 reconstructs per-bin stats via Lagrange coefficients.
// ---------------------------------------------------------------------------
__global__ __launch_bounds__(256) void bin_finalize_kernel(
    const float* __restrict__ ws, const float* __restrict__ alpha,
    float* __restrict__ out, int nrows, int ntot)
{
  const int tx   = threadIdx.x;
  const int wv   = tx >> 5;        // wave id (0..7), wave32
  const int lane = tx & 31;
  const int n    = lane & 15;      // column
  const int half = lane >> 4;      // 0: rows r,r+1   1: rows r+2,r+3

  v8f acc = {};                    // C starts at zero
  v2f ones; ones[0] = 1.0f; ones[1] = 1.0f;

  const int rows_per_wave = nrows >> 3;   // 8 waves
  const int rbeg = wv * rows_per_wave;
  const int rend = rbeg + rows_per_wave;

  for (int r = rbeg; r < rend; r += 4) {
    const int r0 = r + 2 * half;
    v2f bm;
    bm[0] = ws[(r0    ) * 16 + n];
    bm[1] = ws[(r0 + 1) * 16 + n];
    // D = ones(16x4) * B(4x16) + C  -> column sums of 4 partial rows
    acc = __builtin_amdgcn_wmma_f32_16x16x4_f32(
        false, ones, false, bm, (short)0, acc, false, false);
  }

  __shared__ float sh[8][16];
  if (lane < 16) sh[wv][lane] = acc[0];   // D row M=0 lives in VGPR0, lanes 0-15
  __syncthreads();

  __shared__ float col[16];
  if (tx < 16) {
    float s = 0.0f;
#pragma unroll
    for (int w = 0; w < 8; ++w) s += sh[w][tx];
    col[tx] = s;
  }
  __syncthreads();

  if (tx == 0) {
    const double M1 = col[0],  M2 = col[1],  M3 = col[2];
    const double W0 = col[3],  W1 = col[4],  W2 = col[5],  W3 = col[6];
    const double Q0 = col[7],  Q1 = col[8],  Q2 = col[9],  Q3 = col[10];
    const double SA = col[11];
    const double a  = (double)alpha[0];
    const double N  = (double)ntot;

    const double Mv[4] = {N,  M1, M2, M3};
    const double Wv[4] = {W0, W1, W2, W3};
    const double Qv[4] = {Q0, Q1, Q2, Q3};

    // Lagrange indicator coefficients: ind_{b==v}(b) = sum_p C[v][p] * b^p
    // over support b in {-2,-1,0,1}
    const double C[4][4] = {
      {0.0, 1.0/6.0,  0.0, -1.0/6.0},   // v = -2
      {0.0, -1.0,     0.5,  0.5    },   // v = -1
      {1.0, 0.5,     -1.0, -0.5    },   // v =  0
      {0.0, 1.0/3.0,  0.5,  1.0/6.0},   // v =  1
    };

    double tm = 0.0, tv = 0.0;
#pragma unroll
    for (int j = 0; j < 4; ++j) {
      double cnt = 0.0, S = 0.0, Q = 0.0;
#pragma unroll
      for (int p = 0; p < 4; ++p) {
        cnt += C[j][p] * Mv[p];
        S   += C[j][p] * Wv[p];
        Q   += C[j][p] * Qv[p];
      }
      const double lvl  = (double)(j - 2) * a;
      const double safe = cnt > 1.0 ? cnt : 1.0;
      const double mean = S / safe;
      const double var  = Q / safe - mean * mean;
      if (cnt > 0.5) { const double e = mean - lvl; tm += e * e; }
      if (cnt > 1.5) { tv += var; }
    }

    // sum d^2 = Q0 - 2a*W1 + a^2*M2  (d = w - a*b)
    const double sumd2 = Q0 - 2.0 * a * W1 + a * a * M2;

    out[0] = (float)(tm + tv);     // loss
    out[1] = (float)tm;            // total_mse
    out[2] = (float)tv;            // total_var
    out[3] = (float)(sumd2 / N);   // quantization_mse
    out[4] = (float)(SA / N);      // mean_distance
  }
}

// ---------------------------------------------------------------------------
extern "C" void kernel_launch(void* const* d_in, const int* in_sizes, int n_in,
                              void* d_out, int out_size, void* d_ws, size_t ws_size,
                              hipStream_t stream) {
  const v4f*   w4    = (const v4f*)d_in[0];
  const float* alpha = (const float*)d_in[1];
  float*       out   = (float*)d_out;
  float*       ws    = (float*)d_ws;

  const int n  = in_sizes[0];      // 4096*16384 = 67,108,864
  const int n4 = n / 4;

  bin_moments_kernel<<<NBLK, TPB, 0, stream>>>(w4, alpha, ws, n4);
  bin_finalize_kernel<<<1, 256, 0, stream>>>(ws, alpha, out, NBLK, n);
}